// YOLOLoss_21895743275785
// MI455X (gfx1250) — compile-verified
//
#include <hip/hip_runtime.h>
#include <math.h>

#define GRID_N   20
#define NCELLS   400          // 20*20
#define N_ANCH   5
#define NCH      25           // 5 anchors * 5 fields
#define INP_F    640.0f
#define RATIO_F  32.0f        // 640/20
#define BLOCK    256

// 16-byte GCC-style int vector: matches the builtin's parameter pointee type
// exactly as printed by clang ('__vector_size__(4 * sizeof(int)) int').
typedef int v4i __attribute__((vector_size(16)));
typedef __attribute__((address_space(1))) v4i gv4i;   // global (printed __device__)
typedef __attribute__((address_space(3))) v4i lv4i;   // LDS    (shared)

// ---- CDNA5 async global -> LDS copy (128-bit), tracked by ASYNCcnt ----
__device__ __forceinline__ void async_copy_b128(const float* g, float* l) {
#if __has_builtin(__builtin_amdgcn_global_load_async_to_lds_b128)
    __builtin_amdgcn_global_load_async_to_lds_b128(
        (gv4i*)g, (lv4i*)l, /*imm offset*/0, /*cpol*/0);
#elif defined(__gfx1250__)
    // Low 32 bits of a generic pointer into LDS are the wave-relative LDS
    // byte address (flat-aperture encoding), which is what VDST wants.
    unsigned lds_off = (unsigned)(unsigned long long)(void*)l;
    asm volatile("global_load_async_to_lds_b128 %0, %1, off"
                 :: "v"(lds_off), "v"(g) : "memory");
#else
    // host / non-gfx1250 fallback: plain 128-bit copy
    *(float4*)l = *(const float4*)g;
#endif
}

__device__ __forceinline__ void wait_async0() {
#if __has_builtin(__builtin_amdgcn_s_wait_asynccnt)
    __builtin_amdgcn_s_wait_asynccnt(0);
#elif defined(__gfx1250__)
    asm volatile("s_wait_asynccnt 0" ::: "memory");
#endif
}

__device__ __forceinline__ float sigmoid_f(float v) {
    return 1.0f / (1.0f + __expf(-v));
}

// One block per image b. Accumulators:
//  acc0 = obj coord loss sum   (xy sq err + sqrt-wh sq err)  -> /(2B)
//  acc1 = obj conf loss sum    ((conf - iou)^2)              -> /B
//  acc2 = noobj coord sum                                     -> /(B*399*4)
//  acc3 = noobj conf sum                                      -> /(B*399)
__global__ __launch_bounds__(BLOCK)
void yolo_loss_main(const float* __restrict__ pred,
                    const float* __restrict__ label,
                    const float* __restrict__ anchors,
                    float* __restrict__ partial, int B) {
    __shared__ __align__(16) float sp[NCH * NCELLS];  // 40,000 B per-image slice
    __shared__ float rbuf[4][BLOCK];                  // reduction scratch

    const int b   = blockIdx.x;
    const int tid = threadIdx.x;

    // ---- stage pred_[b] (25x400 fp32) into LDS with async b128 DMA ----
    const float* gsrc = pred + (size_t)b * (NCH * NCELLS);
    #pragma unroll 2
    for (int chunk = tid; chunk < (NCH * NCELLS) / 4; chunk += BLOCK) {
        async_copy_b128(gsrc + chunk * 4, &sp[chunk * 4]);
    }
    wait_async0();
    __syncthreads();

    // per-image constants (block-uniform -> scalar loads)
    const float lx = label[b * 4 + 0];
    const float ly = label[b * 4 + 1];
    const float lw = label[b * 4 + 2];
    const float lh = label[b * 4 + 3];
    const int idx  = (int)floorf(lx * (float)GRID_N) * GRID_N
                   + (int)floorf(ly * (float)GRID_N);
    const float A0 = anchors[0] * INP_F;   // anchors[0][0] * 640
    const float A1 = anchors[1] * INP_F;   // anchors[0][1] * 640

    float a_oc = 0.f, a_of = 0.f, a_nc = 0.f, a_nf = 0.f;

    for (int c = tid; c < NCELLS; c += BLOCK) {
        float s0 = 0.f, s1 = 0.f, s2 = 0.f, s3 = 0.f, sc = 0.f;
        #pragma unroll
        for (int a = 0; a < N_ANCH; ++a) {
            s0 += sp[(a * 5 + 0) * NCELLS + c];
            s1 += sp[(a * 5 + 1) * NCELLS + c];
            s2 += sp[(a * 5 + 2) * NCELLS + c];
            s3 += sp[(a * 5 + 3) * NCELLS + c];
            sc += sigmoid_f(sp[(a * 5 + 4) * NCELLS + c]);
        }
        const float x    = sigmoid_f(s0 * 0.2f);
        const float y    = sigmoid_f(s1 * 0.2f);
        const float w    = __expf(s2 * 0.2f) * A0;
        const float h    = __expf(s3 * 0.2f) * A1;
        const float conf = sc * 0.2f;

        if (c != idx) {
            // noobj: target agrid = [0.5, 0.5, A0, A1]
            const float dx = x - 0.5f, dy = y - 0.5f;
            const float dw = w - A0,   dh = h - A1;
            a_nc += dx * dx + dy * dy + dw * dw + dh * dh;
            a_nf += conf * conf;
        } else {
            // object terms at the gathered cell
            const float cxl = lx * (float)GRID_N;
            const float cyl = ly * (float)GRID_N;
            const float wl  = lw * INP_F;
            const float hl  = lh * INP_F;

            const float ex = x - cxl, ey = y - cyl;
            const float ew = sqrtf(w) - sqrtf(wl);
            const float eh = sqrtf(h) - sqrtf(hl);
            a_oc += ex * ex + ey * ey + ew * ew + eh * eh;

            // IoU between label box and predicted box (xyxy, clipped to image)
            const float lx1 = fmaxf(cxl * RATIO_F - wl * 0.5f, 0.f);
            const float ly1 = fmaxf(cyl * RATIO_F - hl * 0.5f, 0.f);
            const float lx2 = fminf(cxl * RATIO_F + wl * 0.5f, INP_F);
            const float ly2 = fminf(cyl * RATIO_F + hl * 0.5f, INP_F);
            const float px1 = fmaxf(x * RATIO_F - w * 0.5f, 0.f);
            const float py1 = fmaxf(y * RATIO_F - h * 0.5f, 0.f);
            const float px2 = fminf(x * RATIO_F + w * 0.5f, INP_F);
            const float py2 = fminf(y * RATIO_F + h * 0.5f, INP_F);
            const float iw = fmaxf(fminf(lx2, px2) - fmaxf(lx1, px1), 0.f);
            const float ih = fmaxf(fminf(ly2, py2) - fmaxf(ly1, py1), 0.f);
            const float inter = iw * ih;
            const float iou = inter / (wl * hl + w * h - inter);
            const float ec = conf - iou;
            a_of += ec * ec;
        }
    }

    // deterministic in-block tree reduction (no atomics)
    rbuf[0][tid] = a_oc;
    rbuf[1][tid] = a_of;
    rbuf[2][tid] = a_nc;
    rbuf[3][tid] = a_nf;
    __syncthreads();
    for (int s = BLOCK / 2; s > 0; s >>= 1) {
        if (tid < s) {
            rbuf[0][tid] += rbuf[0][tid + s];
            rbuf[1][tid] += rbuf[1][tid + s];
            rbuf[2][tid] += rbuf[2][tid + s];
            rbuf[3][tid] += rbuf[3][tid + s];
        }
        __syncthreads();
    }
    if (tid == 0) {
        partial[0 * B + b] = rbuf[0][0];
        partial[1 * B + b] = rbuf[1][0];
        partial[2 * B + b] = rbuf[2][0];
        partial[3 * B + b] = rbuf[3][0];
    }
}

__global__ __launch_bounds__(BLOCK)
void yolo_loss_final(const float* __restrict__ partial,
                     float* __restrict__ out, int B) {
    __shared__ float rbuf[4][BLOCK];
    const int tid = threadIdx.x;
    float s0 = 0.f, s1 = 0.f, s2 = 0.f, s3 = 0.f;
    for (int i = tid; i < B; i += BLOCK) {
        s0 += partial[0 * B + i];
        s1 += partial[1 * B + i];
        s2 += partial[2 * B + i];
        s3 += partial[3 * B + i];
    }
    rbuf[0][tid] = s0; rbuf[1][tid] = s1; rbuf[2][tid] = s2; rbuf[3][tid] = s3;
    __syncthreads();
    for (int s = BLOCK / 2; s > 0; s >>= 1) {
        if (tid < s) {
            rbuf[0][tid] += rbuf[0][tid + s];
            rbuf[1][tid] += rbuf[1][tid + s];
            rbuf[2][tid] += rbuf[2][tid + s];
            rbuf[3][tid] += rbuf[3][tid + s];
        }
        __syncthreads();
    }
    if (tid == 0) {
        const float Bf      = (float)B;
        const float n_noobj = Bf * (float)(NCELLS - 1);
        const float coor_obj   = rbuf[0][0] / (2.0f * Bf);
        const float conf_obj   = rbuf[1][0] / Bf;
        const float coor_noobj = rbuf[2][0] / (n_noobj * 4.0f);
        const float conf_noobj = rbuf[3][0] / n_noobj;
        // weights: L_COOR_OBJ=2, L_COOR_NOOBJ=1, L_CONF_OBJ=5, L_CONF_NOOBJ=1
        out[0] = 2.0f * coor_obj + coor_noobj + 5.0f * conf_obj + conf_noobj;
    }
}

extern "C" void kernel_launch(void* const* d_in, const int* in_sizes, int n_in,
                              void* d_out, int out_size, void* d_ws, size_t ws_size,
                              hipStream_t stream) {
    const float* pred    = (const float*)d_in[0];   // (B, 25, 20, 20) fp32
    const float* label   = (const float*)d_in[1];   // (B, 4) fp32
    const float* anchors = (const float*)d_in[2];   // (5, 2) fp32
    float*       out     = (float*)d_out;           // scalar fp32

    const int B = in_sizes[0] / (NCH * NCELLS);     // 8192
    float* partial = (float*)d_ws;                  // 4*B floats = 128 KB

    yolo_loss_main<<<B, BLOCK, 0, stream>>>(pred, label, anchors, partial, B);
    yolo_loss_final<<<1, BLOCK, 0, stream>>>(partial, out, B);
}